// Codebook_2929167696356
// MI455X (gfx1250) — compile-verified
//
#include <hip/hip_runtime.h>

// VQ-VAE codebook nearest-neighbor + gather for MI455X (gfx1250, wave32).
//
// out[n,:] = weight[argmin_k (|w_k|^2 - 2 z_n . w_k), :]
//
// v2: weight tiles staged to LDS via CDNA5 async global->LDS DMA
// (global_load_async_to_lds_b128, ASYNCcnt), double-buffered, row pitch 260
// floats so B-fragment ds_load_b64 reads are 64-bank conflict-free.
// A-fragments remain register-resident (64 x v2f per lane).

typedef __attribute__((ext_vector_type(2))) float v2f;
typedef __attribute__((ext_vector_type(8))) float v8f;

#define EMB_D   256
#define NUM_K   1024
#define KSTEPS  (EMB_D / 4)   // 64 wmma steps per 16x16 tile
#define KTILES  (NUM_K / 16)  // 64 code tiles
#define LDSP    260           // LDS row pitch in floats (260 % 64 == 4)

// ---------------------------------------------------------------------------
// Kernel 1: wnorm[k] = sum_d w[k,d]^2   (one wave per codebook row)
// ---------------------------------------------------------------------------
__global__ __launch_bounds__(32) void vq_wnorm_kernel(const float* __restrict__ w,
                                                      float* __restrict__ wnorm) {
    const int row  = blockIdx.x;
    const int lane = threadIdx.x;
    const float* p = w + (size_t)row * EMB_D;
    float s = 0.0f;
#pragma unroll
    for (int i = 0; i < EMB_D / 32; ++i) {
        float v = p[lane + 32 * i];
        s = fmaf(v, v, s);
    }
#pragma unroll
    for (int off = 16; off >= 1; off >>= 1)
        s += __shfl_xor(s, off, 32);
    if (lane == 0) wnorm[row] = s;
}

// ---------------------------------------------------------------------------
// Async-stage one 16x256 weight tile into an LDS buffer (pitch LDSP).
// 256 threads x 4 x b128: each instruction moves a coalesced 512B per wave.
// Completion tracked by ASYNCcnt.
// ---------------------------------------------------------------------------
__device__ __forceinline__ void stage_tile(const float* __restrict__ w, int t,
                                           float* ldsbuf, int tid) {
#pragma unroll
    for (int v = 0; v < 4; ++v) {
        const int flat = v * 1024 + tid * 4;   // element index within 16x256 tile
        const int row  = flat >> 8;            // 0..15 (code within tile)
        const int col  = flat & 255;           // 0..255 (depth)
        const uint32_t lds_off = (uint32_t)(uintptr_t)(ldsbuf + row * LDSP + col);
        const float* g = w + (size_t)(t * 16 + row) * EMB_D + col;
        asm volatile("global_load_async_to_lds_b128 %0, %1, off"
                     :: "v"(lds_off), "v"(g)
                     : "memory");
    }
}

// ---------------------------------------------------------------------------
// Kernel 2: fused distance-matmul + argmin + gather.
//   256 threads = 8 waves; each wave owns a 16-row M-tile (block owns 128 rows).
//   All 8 waves share the LDS-staged weight tile (8x cut in cache traffic).
// WMMA f32 16x16x4 operand layouts (ISA 7.12.2):
//   A 16x4 : lanes 0-15 rows M=0..15 hold K=0,1 (VGPR0,1); lanes 16-31 hold K=2,3
//   B 4x16 : VGPR0 lanes0-15 = row K0, lanes16-31 = K2; VGPR1 = K1 / K3
//   C 16x16: VGPR j, lane l -> M = j + 8*(l>=16), N = l & 15
// ---------------------------------------------------------------------------
__global__ __launch_bounds__(256) void vq_argmin_gather_kernel(
    const float* __restrict__ z, const float* __restrict__ w,
    const float* __restrict__ wnorm, float* __restrict__ out) {

    __shared__ __align__(16) float ldsB[2][16 * LDSP];  // double buffer, 2x16.6KB

    const int tid   = threadIdx.x;
    const int lane  = tid & 31;
    const int wave  = tid >> 5;
    const int half  = lane >> 4;   // 0: K pair {0,1}; 1: K pair {2,3}
    const int l16   = lane & 15;   // A: row within tile; B/C: column (code)
    const int mbase = blockIdx.x * 128 + wave * 16;

    // --- A fragments: entire 16x256 tile register-resident (128 VGPRs/lane).
    // a[s] holds z[mbase+l16][4s + 2*half + {0,1}]
    v2f a[KSTEPS];
    {
        const float* zrow = z + (size_t)(mbase + l16) * EMB_D + 2 * half;
#pragma unroll
        for (int s = 0; s < KSTEPS; ++s)
            a[s] = *(const v2f*)(zrow + 4 * s);
    }

    float bestv[8];
    int   besti[8];
#pragma unroll
    for (int j = 0; j < 8; ++j) { bestv[j] = 3.4e38f; besti[j] = 0; }

    // --- Prologue: stage tile 0 into buffer 0.
    stage_tile(w, 0, ldsB[0], tid);

    // --- Sweep codebook in 16-code tiles, double-buffered.
    for (int t = 0; t < KTILES; ++t) {
        const int buf = t & 1;

        // My async stores for tile t are done; barrier makes them visible to
        // all waves AND guarantees everyone finished reading buffer buf^1
        // (used by tile t-1), so it is safe to overwrite below.
        asm volatile("s_wait_asynccnt 0x0" ::: "memory");
        __syncthreads();

        if (t + 1 < KTILES)
            stage_tile(w, t + 1, ldsB[buf ^ 1], tid);

        // B fragments from LDS: lanes hit 64 distinct banks (pitch 260).
        const float* bb = &ldsB[buf][l16 * LDSP + 2 * half];
        v8f c = {};
#pragma unroll
        for (int s = 0; s < KSTEPS; ++s) {
            v2f b = *(const v2f*)(bb + 4 * s);
            c = __builtin_amdgcn_wmma_f32_16x16x4_f32(
                    /*neg_a=*/false, a[s], /*neg_b=*/false, b,
                    /*c_mod=*/(short)0, c, /*reuse_a=*/false, /*reuse_b=*/false);
        }

        const int  code = t * 16 + l16;
        const float wn  = wnorm[code];
#pragma unroll
        for (int j = 0; j < 8; ++j) {
            float d = fmaf(-2.0f, c[j], wn);  // |x|^2 constant per row: dropped
            if (d < bestv[j]) { bestv[j] = d; besti[j] = code; }
        }
    }

    // --- Argmin across the 16 lanes of each half (lanes share rows M=j+8*half).
#pragma unroll
    for (int off = 8; off >= 1; off >>= 1) {
#pragma unroll
        for (int j = 0; j < 8; ++j) {
            float ov = __shfl_xor(bestv[j], off, 32);
            int   oi = __shfl_xor(besti[j], off, 32);
            if (ov < bestv[j] || (ov == bestv[j] && oi < besti[j])) {
                bestv[j] = ov; besti[j] = oi;
            }
        }
    }

    // --- Gather winning codebook rows; 16 lanes/half x 16 floats = 256/row.
#pragma unroll
    for (int j = 0; j < 8; ++j) {
        const int row = mbase + half * 8 + j;
        const float* src = w   + (size_t)besti[j] * EMB_D + l16 * 16;
        float*       dst = out + (size_t)row      * EMB_D + l16 * 16;
#pragma unroll
        for (int v = 0; v < 4; ++v)
            ((float4*)dst)[v] = ((const float4*)src)[v];
    }
}

// ---------------------------------------------------------------------------
extern "C" void kernel_launch(void* const* d_in, const int* in_sizes, int n_in,
                              void* d_out, int out_size, void* d_ws, size_t ws_size,
                              hipStream_t stream) {
    const float* z   = (const float*)d_in[0];   // [32*32*32, 256] = [32768, 256]
    const float* w   = (const float*)d_in[1];   // [1024, 256]
    float* out       = (float*)d_out;           // [32768, 256]
    float* wnorm     = (float*)d_ws;            // 1024 floats of scratch

    const int n_rows = in_sizes[0] / EMB_D;     // 32768 (multiple of 128)

    vq_wnorm_kernel<<<NUM_K, 32, 0, stream>>>(w, wnorm);
    vq_argmin_gather_kernel<<<n_rows / 128, 256, 0, stream>>>(z, w, wnorm, out);
}